// StabilityLossCoP_15101105013423
// MI455X (gfx1250) — compile-verified
//
#include <hip/hip_runtime.h>
#include <math.h>

#define B_      256
#define V_      6890
#define F_      13776
#define NHD_    20000
#define NPARTS  10
#define PY      120
#define PX      40
#define NPIX    (PY * PX)
#define NT      256
#define NWAVES  (NT / 32)

typedef __attribute__((ext_vector_type(2))) float v2f;
typedef __attribute__((ext_vector_type(8))) float v8f;

// Exact fp32 sum of one value across all 32 lanes of a wave, broadcast to
// every lane, using V_WMMA_F32_16X16X4_F32 with an all-ones B operand.
// Relies only on the documented A layout (lane m / m+16 hold K=0,1 / K=2,3)
// and documented C/D layout (VGPR r, lane half -> row r or r+8).
// Stage 1: D[m][n] = x_m + x_{m+16}; per-lane sum of the 8 D registers gives
// S_low (lanes 0-15) / S_high (lanes 16-31). Stage 2 sums those two halves.
__device__ __forceinline__ float wave_sum_wmma(float x) {
  v2f a;    a[0] = x;    a[1] = 0.0f;
  v2f ones; ones[0] = 1.0f; ones[1] = 1.0f;
  v8f c = {};
  v8f d = __builtin_amdgcn_wmma_f32_16x16x4_f32(false, a, false, ones,
                                                (short)0, c, false, false);
  float partial = d[0] + d[1] + d[2] + d[3] + d[4] + d[5] + d[6] + d[7];
  v2f a2; a2[0] = partial; a2[1] = 0.0f;
  v8f d2 = __builtin_amdgcn_wmma_f32_16x16x4_f32(false, a2, false, ones,
                                                 (short)0, c, false, false);
  return d2[0];  // full 32-lane total, identical in every lane
}

// Block-wide sum of N values per thread; results broadcast back into vals[].
// Must be called from uniform control flow (EXEC all ones for WMMA).
template <int N>
__device__ __forceinline__ void block_sum_n(float* vals, float* scratch) {
  const int lane = threadIdx.x & 31;
  const int wid  = threadIdx.x >> 5;
  float w[N];
#pragma unroll
  for (int j = 0; j < N; ++j) w[j] = wave_sum_wmma(vals[j]);
  if (lane == 0) {
#pragma unroll
    for (int j = 0; j < N; ++j) scratch[wid * N + j] = w[j];
  }
  __syncthreads();
#pragma unroll
  for (int j = 0; j < N; ++j) {
    float s = 0.0f;
#pragma unroll
    for (int i = 0; i < NWAVES; ++i) s += scratch[i * N + j];
    vals[j] = s;
  }
  __syncthreads();
}

// One block per batch. Phase 1: per-part tet volumes. Phase 2: barycentric
// samples -> com / cop moments. Writes com_mat and cop_mat_ipman to d_out,
// raw com (x,y) to ws_com for the loss kernel.
__global__ __launch_bounds__(NT) void fused_body_kernel(
    const float* __restrict__ vertices, const float* __restrict__ bary_w,
    const int* __restrict__ faces, const int* __restrict__ face_sample,
    const int* __restrict__ face_part, float* __restrict__ d_out,
    float* __restrict__ ws_com) {
  const int b   = blockIdx.x;
  const int tid = threadIdx.x;
  const float* __restrict__ vb = vertices + (size_t)b * (V_ * 3);

  __shared__ float s_scratch[NWAVES * NPARTS];
  __shared__ float s_partvol[NPARTS];

  // ---- phase 1: per-part signed tet volumes ----
  float acc[NPARTS];
#pragma unroll
  for (int p = 0; p < NPARTS; ++p) acc[p] = 0.0f;

  for (int f = tid; f < F_; f += NT) {
    const int i0 = faces[3 * f + 0];
    const int i1 = faces[3 * f + 1];
    const int i2 = faces[3 * f + 2];
    const float ax = vb[3 * i0 + 0], ay = vb[3 * i0 + 1], az = vb[3 * i0 + 2];
    const float bx = vb[3 * i1 + 0], by = vb[3 * i1 + 1], bz = vb[3 * i1 + 2];
    const float cx = vb[3 * i2 + 0], cy = vb[3 * i2 + 1], cz = vb[3 * i2 + 2];
    const float crx = by * cz - bz * cy;
    const float cry = bz * cx - bx * cz;
    const float crz = bx * cy - by * cx;
    const float tet = (ax * crx + ay * cry + az * crz) * (1.0f / 6.0f);
    const int p = face_part[f];
#pragma unroll
    for (int q = 0; q < NPARTS; ++q) acc[q] += (q == p) ? tet : 0.0f;
  }
  block_sum_n<NPARTS>(acc, s_scratch);
  if (tid == 0) {
#pragma unroll
    for (int p = 0; p < NPARTS; ++p) s_partvol[p] = acc[p];
  }
  __syncthreads();

  // ---- phase 2: barycentric samples -> com / cop moments ----
  // s[0..2]=sum vhd*vol  s[3]=sum vol  s[4..6]=sum vhd*pw  s[7]=sum pw
  float s[8];
#pragma unroll
  for (int j = 0; j < 8; ++j) s[j] = 0.0f;

  for (int n = tid; n < NHD_; n += NT) {
    const int fidx = face_sample[n];
    const int i0 = faces[3 * fidx + 0];
    const int i1 = faces[3 * fidx + 1];
    const int i2 = faces[3 * fidx + 2];
    const float w0 = bary_w[3 * n + 0];
    const float w1 = bary_w[3 * n + 1];
    const float w2 = bary_w[3 * n + 2];
    const float hx = w0 * vb[3 * i0 + 0] + w1 * vb[3 * i1 + 0] + w2 * vb[3 * i2 + 0];
    const float hy = w0 * vb[3 * i0 + 1] + w1 * vb[3 * i1 + 1] + w2 * vb[3 * i2 + 1];
    const float hz = w0 * vb[3 * i0 + 2] + w1 * vb[3 * i1 + 2] + w2 * vb[3 * i2 + 2];
    const float vol = s_partvol[face_part[fidx]];
    const float pw = (hy < 0.0f) ? fmaf(-100.0f, hy, 1.0f) : expf(-10.0f * hy);
    s[0] += hx * vol; s[1] += hy * vol; s[2] += hz * vol; s[3] += vol;
    s[4] += hx * pw;  s[5] += hy * pw;  s[6] += hz * pw;  s[7] += pw;
  }
  block_sum_n<8>(s, s_scratch);

  if (tid == 0) {
    const float com_x = s[0] / s[3];
    const float com_y = s[1] / s[3];
    const float denom = s[7] + 1e-6f;
    const float cop_x = s[4] / denom;
    const float cop_y = s[5] / denom;
    ws_com[2 * b + 0] = com_x;
    ws_com[2 * b + 1] = com_y;
    d_out[1 + 2 * b + 0] = com_x * 48.0f;
    d_out[1 + 2 * b + 1] = (59.0f / 33.0f - com_y) * 33.0f;
    d_out[1 + 2 * B_ + 2 * b + 0] = (cop_x + 18.0f / 78.74f) * 78.74f;
    d_out[1 + 2 * B_ + 2 * b + 1] = (86.0f / 78.74f - cop_y) * 78.74f;
  }
}

// One block per batch: pressure-grid centroid -> raw cop_xy in workspace.
__global__ __launch_bounds__(NT) void pressure_cop_kernel(
    const float* __restrict__ pressure, float* __restrict__ ws_cop) {
  const int b   = blockIdx.x;
  const int tid = threadIdx.x;
  const float* __restrict__ pb = pressure + (size_t)b * NPIX;

  __shared__ float s_scratch[NWAVES * 3];
  float s[3] = {0.0f, 0.0f, 0.0f};  // total, x-weighted, y-weighted
  for (int i = tid; i < NPIX; i += NT) {
    const float p = pb[i];
    s[0] += p;
    s[1] += p * (float)(i % PX);
    s[2] += p * (float)(i / PX);
  }
  block_sum_n<3>(s, s_scratch);
  if (tid == 0) {
    const float t   = (s[0] == 0.0f) ? 1e-8f : s[0];
    const float x_c = s[1] / t;
    const float y_c = s[2] / t;
    ws_cop[2 * b + 0] = 0.01f * (x_c - 18.0f) / 0.7874f;
    ws_cop[2 * b + 1] = 0.01f * (86.0f - y_c) / 0.7874f;
  }
}

// Single block, 256 threads (B == 256): mean 2D distance -> d_out[0].
__global__ __launch_bounds__(NT) void loss_kernel(
    const float* __restrict__ ws_com, const float* __restrict__ ws_cop,
    float* __restrict__ d_out) {
  const int tid = threadIdx.x;
  __shared__ float s_scratch[NWAVES];
  const float dx = ws_com[2 * tid + 0] - ws_cop[2 * tid + 0];
  const float dy = ws_com[2 * tid + 1] - ws_cop[2 * tid + 1];
  float v[1];
  v[0] = sqrtf(dx * dx + dy * dy);
  block_sum_n<1>(v, s_scratch);
  if (tid == 0) d_out[0] = v[0] * (1.0f / (float)B_);
}

extern "C" void kernel_launch(void* const* d_in, const int* in_sizes, int n_in,
                              void* d_out, int out_size, void* d_ws,
                              size_t ws_size, hipStream_t stream) {
  (void)in_sizes; (void)n_in; (void)out_size; (void)ws_size;
  const float* vertices    = (const float*)d_in[0];  // (256, 6890, 3)
  const float* pressure    = (const float*)d_in[1];  // (256, 120, 40)
  const float* bary_w      = (const float*)d_in[2];  // (20000, 3)
  const int*   faces       = (const int*)d_in[3];    // (13776, 3)
  const int*   face_sample = (const int*)d_in[4];    // (20000,)
  const int*   face_part   = (const int*)d_in[5];    // (13776,)

  float* out    = (float*)d_out;
  float* ws     = (float*)d_ws;
  float* ws_com = ws;            // 512 floats: raw com (x,y) per batch
  float* ws_cop = ws + 2 * B_;   // 512 floats: raw cop_xy per batch

  fused_body_kernel<<<B_, NT, 0, stream>>>(vertices, bary_w, faces,
                                           face_sample, face_part, out,
                                           ws_com);
  pressure_cop_kernel<<<B_, NT, 0, stream>>>(pressure, ws_cop);
  loss_kernel<<<1, NT, 0, stream>>>(ws_com, ws_cop, out);
}